// VocalEncoder_23218593202910
// MI455X (gfx1250) — compile-verified
//
#include <hip/hip_runtime.h>
#include <cstdint>

typedef float v4f __attribute__((ext_vector_type(4)));

#define N_PITCH   576
#define D_MODEL   128
#define NTOK      (64 * 4096)            // B*S = 262144 tokens
#define WT_MAIN   (N_PITCH * D_MODEL)    // 73728 floats (transposed table)
#define WT_TOTAL  (WT_MAIN + 2 * D_MODEL) // + w_gain[128] + bias[128]
#define SW4_N     (WT_MAIN / 4)          // 18432 v4f in LDS table
#define WGB4_N    (2 * D_MODEL / 4)      // 64 v4f (w_gain ++ bias)

// ---------------------------------------------------------------------------
// Prep: Wt[p*128+d] = W[d*577+p];  Wt[73728+d] = W[d*577+576];  Wt[73856+d]=b[d]
// ---------------------------------------------------------------------------
__global__ void vocal_prep_kernel(const float* __restrict__ W,
                                  const float* __restrict__ bias,
                                  float* __restrict__ Wt) {
    int idx = blockIdx.x * blockDim.x + threadIdx.x;
    if (idx < WT_MAIN) {
        int p = idx >> 7;            // /128
        int d = idx & 127;
        Wt[idx] = W[d * (N_PITCH + 1) + p];
    } else if (idx < WT_MAIN + D_MODEL) {
        int d = idx - WT_MAIN;
        Wt[idx] = W[d * (N_PITCH + 1) + N_PITCH];   // gain column
    } else if (idx < WT_TOTAL) {
        int d = idx - (WT_MAIN + D_MODEL);
        Wt[idx] = bias[d];
    }
}

// ---------------------------------------------------------------------------
// Main: stage table in LDS via async-to-LDS DMA, then per-token
//   out[t, 4l..4l+3] = Wt_row(p)[..] + g * wg[..] + b[..]
// one wave per token, lane l covers 16 bytes -> 512B coalesced NT store/token.
// ---------------------------------------------------------------------------
__global__ void __launch_bounds__(1024, 1)
vocal_main_kernel(const int*   __restrict__ pitch,
                  const float* __restrict__ gain,
                  const float* __restrict__ Wt,   // WT_TOTAL floats in d_ws
                  float*       __restrict__ out) {
    __shared__ v4f sW4[SW4_N];     // 294912 B: 576 rows x 128 floats
    __shared__ v4f sWgB4[WGB4_N];  //   1024 B: w_gain[128] ++ bias[128]

    const v4f* __restrict__ Wt4 = (const v4f*)Wt;

    // ---- async fill of LDS (CDNA5 GLOBAL_LOAD_ASYNC_TO_LDS_B128, ASYNCcnt) ----
    for (int i = threadIdx.x; i < SW4_N; i += blockDim.x) {
        uint32_t laddr = (uint32_t)(uintptr_t)(&sW4[i]);
        uint64_t gaddr = (uint64_t)(uintptr_t)(Wt4 + i);
        asm volatile("global_load_async_to_lds_b128 %0, %1, off"
                     :: "v"(laddr), "v"(gaddr) : "memory");
    }
    if (threadIdx.x < WGB4_N) {
        uint32_t laddr = (uint32_t)(uintptr_t)(&sWgB4[threadIdx.x]);
        uint64_t gaddr = (uint64_t)(uintptr_t)(Wt4 + SW4_N + threadIdx.x);
        asm volatile("global_load_async_to_lds_b128 %0, %1, off"
                     :: "v"(laddr), "v"(gaddr) : "memory");
    }
    asm volatile("s_wait_asynccnt 0x0" ::: "memory");
    __syncthreads();

    const int lane   = threadIdx.x & 31;
    const int wid    = threadIdx.x >> 5;
    const int nwaves = (blockDim.x >> 5) * gridDim.x;
    const int gw     = blockIdx.x * (blockDim.x >> 5) + wid;

    // per-lane slice of w_gain and bias (invariant across tokens)
    const v4f wg4 = sWgB4[lane];
    const v4f b4  = sWgB4[32 + lane];

    v4f* __restrict__ out4 = (v4f*)out;

    // 32 tokens per wave-chunk: one coalesced load of pitch/gain, then
    // broadcast each token's scalars via v_readlane (__shfl).
    const int nchunks = NTOK >> 5;   // 8192
    for (int c = gw; c < nchunks; c += nwaves) {
        const int t0 = c << 5;
        const int   pv = pitch[t0 + lane];
        const float gv = gain [t0 + lane];
        #pragma unroll 8
        for (int k = 0; k < 32; ++k) {
            const int   p = __shfl(pv, k);
            const float g = __shfl(gv, k);
            const v4f w4 = sW4[(p << 5) + lane];        // ds_load_b128
            const v4f o  = w4 + g * wg4 + b4;           // v_fma_f32 x4
            // streaming, write-once output -> non-temporal b128 store
            __builtin_nontemporal_store(o, &out4[((t0 + k) << 5) + lane]);
        }
    }
}

// ---------------------------------------------------------------------------
extern "C" void kernel_launch(void* const* d_in, const int* in_sizes, int n_in,
                              void* d_out, int out_size, void* d_ws, size_t ws_size,
                              hipStream_t stream) {
    const int*   pitch = (const int*)  d_in[0];  // (64, 4096) int32
    const float* gain  = (const float*)d_in[1];  // (64, 4096) f32
    const float* W     = (const float*)d_in[2];  // (128, 577) f32
    const float* bias  = (const float*)d_in[3];  // (128,)     f32
    float*       out   = (float*)d_out;          // (64, 4096, 128) f32
    float*       Wt    = (float*)d_ws;           // WT_TOTAL floats scratch

    // 1) repack W into gather-friendly layout
    {
        int n      = WT_TOTAL;
        int block  = 256;
        int grid   = (n + block - 1) / block;
        vocal_prep_kernel<<<grid, block, 0, stream>>>(W, bias, Wt);
    }

    // 2) persistent main kernel: ~one block per CU (each block owns its LDS copy)
    int dev = 0, ncu = 256;
    (void)hipGetDevice(&dev);
    (void)hipDeviceGetAttribute(&ncu, hipDeviceAttributeMultiprocessorCount, dev);
    if (ncu < 64)   ncu = 64;
    if (ncu > 2048) ncu = 2048;
    vocal_main_kernel<<<ncu, 1024, 0, stream>>>(pitch, gain, Wt, out);
}